// GaiaGridEncoder_69690139345478
// MI455X (gfx1250) — compile-verified
//
#include <hip/hip_runtime.h>
#include <math.h>

// Problem constants (from reference)
#define B_     2
#define C_     64
#define H_     181
#define W_     360
#define N_TOK  (H_ * W_)     // 65160
#define HID    256
#define M_     2562          // icosphere SUBDIV=4 vertex count
#define KNN    8
#define KCOMB  68            // 64 (x feats) + 3 (pos) + 1 (bias row)
#define LN_EPS 1e-5f

typedef __attribute__((ext_vector_type(2))) float v2f;
typedef __attribute__((ext_vector_type(8))) float v8f;

// Native fp32 WMMA: D(16x16) = A(16x4) * B(4x16) + C  — wave32, 2 VGPR A/B frags, 8 VGPR acc.
__device__ __forceinline__ v8f wmma4f32(v2f a, v2f b, v8f c) {
    return __builtin_amdgcn_wmma_f32_16x16x4_f32(false, a, false, b, (short)0, c, false, false);
}

// ---------------------------------------------------------------- token_pos
__global__ void token_pos_kernel(const float* __restrict__ lat,
                                 const float* __restrict__ lon,
                                 float* __restrict__ tp) {
    int n = blockIdx.x * blockDim.x + threadIdx.x;
    if (n >= N_TOK) return;
    int h = n / W_, w = n - h * W_;
    float la = lat[h], lo = lon[w];
    float cl = cosf(la);
    tp[n * 3 + 0] = cl * cosf(lo);
    tp[n * 3 + 1] = cl * sinf(lo);
    tp[n * 3 + 2] = sinf(la);
}

// ------------------------------------------------- combined embed weight [Wt;Wp;bt+bp]
__global__ void wcomb_kernel(const float* __restrict__ Wt, const float* __restrict__ bt,
                             const float* __restrict__ Wp, const float* __restrict__ bp,
                             float* __restrict__ Wcomb) {
    int idx = blockIdx.x * blockDim.x + threadIdx.x;
    if (idx >= KCOMB * HID) return;
    int r = idx / HID, h = idx - r * HID;
    float v;
    if (r < C_)          v = Wt[r * HID + h];
    else if (r < C_ + 3) v = Wp[(r - C_) * HID + h];
    else                 v = bt[h] + bp[h];
    Wcomb[idx] = v;
}

// ---------------------------------------------------------------- pe = mesh_pos@Wp + bp
__global__ void pe_kernel(const float* __restrict__ mesh_pos,
                          const float* __restrict__ Wp, const float* __restrict__ bp,
                          float* __restrict__ pe) {
    int m = blockIdx.x, h = threadIdx.x;
    float px = mesh_pos[m * 3], py = mesh_pos[m * 3 + 1], pz = mesh_pos[m * 3 + 2];
    pe[m * HID + h] = px * Wp[h] + py * Wp[HID + h] + pz * Wp[2 * HID + h] + bp[h];
}

// ------------------- WMMA fp32 GEMM, K=256, N=256: D = A(Mrows x 256) * B + bias
// TB=0: B row-major (256 x 256). TB=1: B transposed access (D = A * B^T).
// Block = 128 threads (4 waves); wave w owns cols [w*64, w*64+64) as 4 x 16x16 tiles.
// Grid.x = ceil(Mrows/16). Branch-free inner loop, 4 independent WMMA chains per wave.
template <int TB>
__global__ void gemm_wmma_f32_k256(const float* __restrict__ A, int lda,
                                   const float* __restrict__ Bm, int ldb,
                                   const float* __restrict__ bias,
                                   float* __restrict__ D, int ldd, int Mrows) {
    const int lane  = threadIdx.x & 31;
    const int wave  = threadIdx.x >> 5;
    const int row   = blockIdx.x * 16 + (lane & 15);
    const int rowc  = (row < Mrows) ? row : (Mrows - 1);   // clamp: OOB rows never stored
    const int khalf = (lane >> 4) * 2;                     // lanes 0-15 -> K+0/1, 16-31 -> K+2/3
    const int col0  = wave * 64 + (lane & 15);
    const float* __restrict__ Arow = A + (size_t)rowc * lda;

    __builtin_prefetch(Bm, 0, 1);                          // global_prefetch_b8
    v8f acc[4] = {};
#pragma unroll 4
    for (int k0 = 0; k0 < 256; k0 += 4) {
        const int ka = k0 + khalf;
        v2f a;
        a.x = Arow[ka];
        a.y = Arow[ka + 1];
#pragma unroll
        for (int t = 0; t < 4; ++t) {
            const int col = col0 + t * 16;
            v2f b;
            if (TB) { b.x = Bm[(size_t)col * ldb + ka]; b.y = Bm[(size_t)col * ldb + ka + 1]; }
            else    { b.x = Bm[(size_t)ka * ldb + col]; b.y = Bm[(size_t)(ka + 1) * ldb + col]; }
            acc[t] = wmma4f32(a, b, acc[t]);
        }
    }
    float bc[4];
#pragma unroll
    for (int t = 0; t < 4; ++t) bc[t] = bias ? bias[col0 + t * 16] : 0.0f;
#pragma unroll
    for (int t = 0; t < 4; ++t) {
#pragma unroll
        for (int v = 0; v < 8; ++v) {
            int r = blockIdx.x * 16 + v + ((lane < 16) ? 0 : 8);
            if (r < Mrows) D[(size_t)r * ldd + col0 + t * 16] = acc[t][v] + bc[t];
        }
    }
}

// ---------------------------------------------------------------- qb[m] = dot(query[m], bk)
__global__ void qb_kernel(const float* __restrict__ query, const float* __restrict__ bk,
                          float* __restrict__ qb) {
    int m = blockIdx.x * blockDim.x + threadIdx.x;
    if (m >= M_) return;
    float s = 0.f;
    for (int h = 0; h < HID; ++h) s += query[m * HID + h] * bk[h];
    qb[m] = s;
}

// ---------------------------------------------------------------- KNN: top-8 nearest tokens
__global__ void knn_kernel(const float* __restrict__ mesh_pos,
                           const float* __restrict__ tp,
                           int* __restrict__ knn_idx, float* __restrict__ knn_dist) {
    __shared__ float cd[128 * KNN];
    __shared__ int   ci[128 * KNN];
    const int m = blockIdx.x, tid = threadIdx.x;
    float mx = mesh_pos[m * 3], my = mesh_pos[m * 3 + 1], mz = mesh_pos[m * 3 + 2];
    float sm = mx * mx + my * my + mz * mz;

    float bd[KNN]; int bi[KNN];
#pragma unroll
    for (int j = 0; j < KNN; ++j) { bd[j] = 3.4e38f; bi[j] = -1; }

    for (int n = tid; n < N_TOK; n += 128) {
        float tx = tp[n * 3], ty = tp[n * 3 + 1], tz = tp[n * 3 + 2];
        float st = tx * tx + ty * ty + tz * tz;
        float d2 = sm + st - 2.0f * (mx * tx + my * ty + mz * tz);
        if (d2 < bd[KNN - 1]) {
            bd[KNN - 1] = d2; bi[KNN - 1] = n;
#pragma unroll
            for (int j = KNN - 1; j > 0; --j) {
                if (bd[j] < bd[j - 1]) {
                    float td = bd[j]; bd[j] = bd[j - 1]; bd[j - 1] = td;
                    int   ti = bi[j]; bi[j] = bi[j - 1]; bi[j - 1] = ti;
                }
            }
        }
    }
#pragma unroll
    for (int j = 0; j < KNN; ++j) { cd[tid * KNN + j] = bd[j]; ci[tid * KNN + j] = bi[j]; }

    // pairwise sorted-merge reduction: 128 sorted 8-lists -> 1 sorted 8-list (7 rounds)
    for (int stride = 64; stride >= 1; stride >>= 1) {
        __syncthreads();
        if (tid < stride) {
            float od[KNN]; int oi[KNN];
            int ia = 0, ib = 0;
            const int a0 = tid * KNN, b0 = (tid + stride) * KNN;
#pragma unroll
            for (int j = 0; j < KNN; ++j) {
                float da = (ia < KNN) ? cd[a0 + ia] : 3.4e38f;
                float db = (ib < KNN) ? cd[b0 + ib] : 3.4e38f;
                if (da <= db) { od[j] = da; oi[j] = ci[a0 + ia]; ++ia; }
                else          { od[j] = db; oi[j] = ci[b0 + ib]; ++ib; }
            }
#pragma unroll
            for (int j = 0; j < KNN; ++j) { cd[a0 + j] = od[j]; ci[a0 + j] = oi[j]; }
        }
    }
    __syncthreads();
    if (tid < KNN) {
        knn_idx[m * KNN + tid]  = ci[tid];
        knn_dist[m * KNN + tid] = sqrtf(fmaxf(cd[tid], 1e-12f));
    }
}

// --------------------- per-vertex: async-gather + WMMA on-the-fly embed of 16 rows
// (2 batches x 8 neighbors), logits via precomputed qk, softmax over k, weighted sum -> wte.
__global__ void attn_kernel(const float* __restrict__ x,
                            const float* __restrict__ tp,
                            const float* __restrict__ Wcomb,
                            const float* __restrict__ qk,
                            const float* __restrict__ qb,
                            const int* __restrict__ knn_idx,
                            const float* __restrict__ knn_dist,
                            float* __restrict__ wte) {
    __shared__ float sA[16 * KCOMB];    // gathered [x | pos | 1] rows
    __shared__ float sT[16 * HID];      // token embeddings (teb)
    __shared__ float sLog[16];
    __shared__ float sAttn[16];
    const int m = blockIdx.x, tid = threadIdx.x;  // 128 threads = 4 waves
    const int lane = tid & 31, wave = tid >> 5;

    // async global->LDS gather of x features: 16 rows x 64 feats = 1024 elems, 8 per thread
    for (int idx = tid; idx < 16 * C_; idx += 128) {
        int r = idx >> 6, c = idx & 63;
        int b = r >> 3, k = r & 7;
        int i = knn_idx[m * KNN + k];
        unsigned long long ga = (unsigned long long)(uintptr_t)&x[(size_t)(b * C_ + c) * N_TOK + i];
        unsigned lds_off = (unsigned)(uintptr_t)&sA[r * KCOMB + c];  // low 32 bits = LDS offset
        asm volatile("global_load_async_to_lds_b32 %0, %1, off"
                     :: "v"(lds_off), "v"(ga) : "memory");
    }
    // pos + bias-row tail: 16 rows x 4 cols, plain LDS stores
    for (int idx = tid; idx < 16 * 4; idx += 128) {
        int r = idx >> 2, c = C_ + (idx & 3);
        int i = knn_idx[m * KNN + (r & 7)];
        sA[r * KCOMB + c] = (c < C_ + 3) ? tp[i * 3 + (c - C_)] : 1.0f;
    }
    asm volatile("s_wait_asynccnt 0x0" ::: "memory");
    __syncthreads();

    // teb = sA(16x68) @ Wcomb(68x256): each wave owns a 16x64 strip, A-frag reused 4x
    const int arow  = lane & 15;
    const int khalf = (lane >> 4) * 2;
    const int colb  = wave * 64 + (lane & 15);
    v8f acc[4] = {};
#pragma unroll
    for (int k0 = 0; k0 < KCOMB; k0 += 4) {
        const int ka = k0 + khalf;
        v2f a;
        a.x = sA[arow * KCOMB + ka];
        a.y = sA[arow * KCOMB + ka + 1];
#pragma unroll
        for (int t = 0; t < 4; ++t) {
            const int col = colb + t * 16;
            v2f b;
            b.x = Wcomb[ka * HID + col];
            b.y = Wcomb[(ka + 1) * HID + col];
            acc[t] = wmma4f32(a, b, acc[t]);
        }
    }
#pragma unroll
    for (int t = 0; t < 4; ++t) {
#pragma unroll
        for (int v = 0; v < 8; ++v) {
            int r = v + ((lane < 16) ? 0 : 8);
            sT[r * HID + colb + t * 16] = acc[t][v];
        }
    }
    __syncthreads();

    if (tid < 16) {                      // logits: teb . (Wk @ q) identity
        int k = tid & 7;
        const float* tr = &sT[tid * HID];
        const float* qv = &qk[m * HID];
        float s = 0.f;
        for (int h = 0; h < HID; ++h) s += tr[h] * qv[h];
        sLog[tid] = (s + qb[m]) * 0.0625f - knn_dist[m * KNN + k];
    }
    __syncthreads();
    if (tid < 2) {                       // softmax over 8 neighbors, per batch
        float mxv = -3.4e38f;
        for (int k = 0; k < 8; ++k) mxv = fmaxf(mxv, sLog[tid * 8 + k]);
        float e[8], sum = 0.f;
        for (int k = 0; k < 8; ++k) { e[k] = expf(sLog[tid * 8 + k] - mxv); sum += e[k]; }
        float inv = 1.0f / sum;
        for (int k = 0; k < 8; ++k) sAttn[tid * 8 + k] = e[k] * inv;
    }
    __syncthreads();
    for (int idx = tid; idx < 2 * HID; idx += 128) {  // wte = sum_k attn_k * teb_k
        int b = idx >> 8, h = idx & 255;
        float s = 0.f;
#pragma unroll
        for (int k = 0; k < 8; ++k) s += sAttn[b * 8 + k] * sT[(b * 8 + k) * HID + h];
        wte[(size_t)(b * M_ + m) * HID + h] = s;
    }
}

// ---------------------------------------------------------------- LayerNorm -> output
__global__ void ln_kernel(const float* __restrict__ mf,
                          const float* __restrict__ gamma, const float* __restrict__ beta,
                          float* __restrict__ out) {
    __shared__ float red[HID];
    int row = blockIdx.x, h = threadIdx.x;
    float v = mf[row * HID + h];
    red[h] = v; __syncthreads();
    for (int s = HID / 2; s > 0; s >>= 1) { if (h < s) red[h] += red[h + s]; __syncthreads(); }
    float mu = red[0] * (1.0f / HID);
    __syncthreads();
    float d = v - mu;
    red[h] = d * d; __syncthreads();
    for (int s = HID / 2; s > 0; s >>= 1) { if (h < s) red[h] += red[h + s]; __syncthreads(); }
    float var = red[0] * (1.0f / HID);
    out[row * HID + h] = d * rsqrtf(var + LN_EPS) * gamma[h] + beta[h];
}

__global__ void meshpos_copy_kernel(const float* __restrict__ mesh_pos,
                                    float* __restrict__ out) {
    int i = blockIdx.x * blockDim.x + threadIdx.x;
    if (i < M_ * 3) out[B_ * M_ * HID + i] = mesh_pos[i];
}

// ================================================================= launch
extern "C" void kernel_launch(void* const* d_in, const int* in_sizes, int n_in,
                              void* d_out, int out_size, void* d_ws, size_t ws_size,
                              hipStream_t stream) {
    (void)in_sizes; (void)n_in; (void)out_size; (void)ws_size;
    const float* x        = (const float*)d_in[0];
    const float* lat      = (const float*)d_in[1];
    const float* lon      = (const float*)d_in[2];
    const float* mesh_pos = (const float*)d_in[3];
    const float* Wt       = (const float*)d_in[4];
    const float* bt       = (const float*)d_in[5];
    const float* Wp       = (const float*)d_in[6];
    const float* bp       = (const float*)d_in[7];
    const float* Wq       = (const float*)d_in[8];
    const float* bq       = (const float*)d_in[9];
    const float* Wk       = (const float*)d_in[10];
    const float* bk       = (const float*)d_in[11];
    const float* Wv       = (const float*)d_in[12];
    const float* bv       = (const float*)d_in[13];
    const float* gamma    = (const float*)d_in[14];
    const float* beta     = (const float*)d_in[15];
    float* out = (float*)d_out;

    float* wsf = (float*)d_ws;
    size_t off = 0;
    auto carve = [&](size_t n) { float* p = wsf + off; off += (n + 63) & ~(size_t)63; return p; };
    float* tp       = carve((size_t)N_TOK * 3);
    float* Wcomb    = carve((size_t)KCOMB * HID);
    float* pe       = carve((size_t)M_ * HID);
    float* query    = carve((size_t)M_ * HID);
    float* qk       = carve((size_t)M_ * HID);
    float* qb       = carve((size_t)M_);
    float* knn_dist = carve((size_t)M_ * KNN);
    float* wte      = carve((size_t)B_ * M_ * HID);
    float* mf       = carve((size_t)B_ * M_ * HID);
    int*   knn_idx  = (int*)carve((size_t)M_ * KNN);

    token_pos_kernel<<<(N_TOK + 255) / 256, 256, 0, stream>>>(lat, lon, tp);
    wcomb_kernel<<<(KCOMB * HID + 255) / 256, 256, 0, stream>>>(Wt, bt, Wp, bp, Wcomb);
    pe_kernel<<<M_, HID, 0, stream>>>(mesh_pos, Wp, bp, pe);

    // query = pe @ Wq + bq   (WMMA fp32)
    gemm_wmma_f32_k256<0><<<(M_ + 15) / 16, 128, 0, stream>>>(
        pe, HID, Wq, HID, bq, query, HID, M_);
    // qk = query @ Wk^T      (WMMA fp32, transposed B)
    gemm_wmma_f32_k256<1><<<(M_ + 15) / 16, 128, 0, stream>>>(
        query, HID, Wk, HID, nullptr, qk, HID, M_);
    qb_kernel<<<(M_ + 127) / 128, 128, 0, stream>>>(query, bk, qb);

    knn_kernel<<<M_, 128, 0, stream>>>(mesh_pos, tp, knn_idx, knn_dist);

    attn_kernel<<<M_, 128, 0, stream>>>(x, tp, Wcomb, qk, qb, knn_idx, knn_dist, wte);

    // mesh_features = wte @ Wv + bv  (WMMA fp32), then LayerNorm into d_out
    gemm_wmma_f32_k256<0><<<(B_ * M_ + 15) / 16, 128, 0, stream>>>(
        wte, HID, Wv, HID, bv, mf, HID, B_ * M_);
    ln_kernel<<<B_ * M_, HID, 0, stream>>>(mf, gamma, beta, out);

    meshpos_copy_kernel<<<(M_ * 3 + 255) / 256, 256, 0, stream>>>(mesh_pos, out);
}